// PythonFFT_53584011985648
// MI455X (gfx1250) — compile-verified
//
#include <hip/hip_runtime.h>
#include <math.h>

// ---------------------------------------------------------------------------
// Batched FFT, N = 4096 = 16*16*16, f32, one workgroup per row.
// Three radix-16 passes; each DFT-16 is a 16x16 matmul done with chained
// V_WMMA_F32_16X16X4_F32. Row lives in LDS (bank-conflict-free padded layout
// g(x)=x+(x>>4), produced directly by the TDM's LDS-padding feature).
// Twiddles come from a 4096-entry f32 table (built once into d_ws, staged
// into LDS by TDM, stored digit-swapped for conflict-free lookups).
// ---------------------------------------------------------------------------

typedef float        v2f  __attribute__((ext_vector_type(2)));
typedef float        v8f  __attribute__((ext_vector_type(8)));
typedef unsigned int v4u  __attribute__((ext_vector_type(4)));
typedef int          v4i  __attribute__((ext_vector_type(4)));
typedef int          v8i  __attribute__((ext_vector_type(8)));

#define FFT_N    4096
#define TWO_PI_F 6.28318530717958647692f

__device__ inline int gmap(int x) { return x + (x >> 4); }          // padded plane index
__device__ inline int rho(int e)  { return (e >> 4) | ((e & 15) << 8); } // twiddle slot

// ---- TDM: 1-D tile (nelem f32) global -> LDS; optional +1dw/16dw padding --
__device__ inline void tdm_load_1d(unsigned long long gaddr, unsigned int lds_off,
                                   int nelem, bool pad)
{
    v4u g0;
    g0[0] = 1u;                                           // count = 1 (valid D#)
    g0[1] = lds_off;                                      // lds_addr (bytes)
    g0[2] = (unsigned int)(gaddr & 0xffffffffull);        // global_addr[31:0]
    g0[3] = (unsigned int)((gaddr >> 32) & 0x01ffffffull) // global_addr[56:32]
          | (2u << 30);                                   // type = 2 ("image")

    v8i g1;
    g1[0] = 0x00020000                                    // wg_mask=0, data_size=2 (4B)
          | (pad ? ((1 << 20) | (3 << 22)) : 0);          // pad_enable, 16dw interval, 1dw pad
    g1[1] = (int)((nelem & 0xffff) << 16);                // tensor_dim0[15:0]
    g1[2] = (int)(((nelem >> 16) & 0xffff) | (1 << 16));  // tensor_dim0 hi | tensor_dim1=1
    g1[3] = (int)((nelem & 0xffff) << 16);                // tile_dim0 = nelem
    g1[4] = 0;                                            // tile_dim1 = tile_dim2 = 0
    g1[5] = nelem;                                        // tensor_dim0_stride
    g1[6] = 0;
    g1[7] = 0;

    v4i g2 = {0, 0, 0, 0};
    v4i g3 = {0, 0, 0, 0};
    v8i g4 = {0, 0, 0, 0, 0, 0, 0, 0};
    __builtin_amdgcn_tensor_load_to_lds(g0, g1, g2, g3, g4, 0);
}

// ---- one radix-16 pass over one column tile owned by this wave ------------
// PASS 0: along n2 (stride 256), no twiddle
// PASS 1: along n1 (stride 16),  twiddle w4096^(16*K*tile)
// PASS 2: along n0 (stride 1),   twiddle w4096^(K*(tile+16*nloc)); out -> lout
template <int PASS>
__device__ inline void fft_pass_tile(float* __restrict__ lre, float* __restrict__ lim,
                                     const v2f* __restrict__ ltw, v2f* __restrict__ lout,
                                     int tile, int lane,
                                     const v2f* wre, const v2f* wim, const v2f* wimn)
{
    const int half = lane >> 4;
    const int nloc = lane & 15;

    v2f bre[4], bim[4];
#pragma unroll
    for (int cc = 0; cc < 4; ++cc) {
#pragma unroll
        for (int j = 0; j < 2; ++j) {
            const int K = 4 * cc + 2 * half + j;
            int idx, e;
            if (PASS == 0) {
                idx = (tile * 16 + nloc) + (K << 8);
                e   = 0;
            } else if (PASS == 1) {
                idx = nloc + (K << 4) + (tile << 8);
                e   = (K * tile) << 4;
            } else {
                idx = K + (nloc << 4) + (tile << 8);
                e   = K * (tile + (nloc << 4));
            }
            const int gi   = gmap(idx);
            const float re = lre[gi];
            const float im = lim[gi];
            if (PASS == 0) {
                bre[cc][j] = re;
                bim[cc][j] = im;
            } else {
                const v2f t = ltw[rho(e)];
                bre[cc][j] = re * t.x - im * t.y;
                bim[cc][j] = re * t.y + im * t.x;
            }
        }
    }

    v8f acc_re = {};
    v8f acc_im = {};
#pragma unroll
    for (int cc = 0; cc < 4; ++cc) {
        // Re += Wre*Dre ; Re += (-Wim)*Dim
        acc_re = __builtin_amdgcn_wmma_f32_16x16x4_f32(false, wre[cc],  false, bre[cc],
                                                       (short)0, acc_re, false, false);
        acc_re = __builtin_amdgcn_wmma_f32_16x16x4_f32(false, wimn[cc], false, bim[cc],
                                                       (short)0, acc_re, false, false);
        // Im += Wre*Dim ; Im += Wim*Dre
        acc_im = __builtin_amdgcn_wmma_f32_16x16x4_f32(false, wre[cc],  false, bim[cc],
                                                       (short)0, acc_im, false, false);
        acc_im = __builtin_amdgcn_wmma_f32_16x16x4_f32(false, wim[cc],  false, bre[cc],
                                                       (short)0, acc_im, false, false);
    }

    // C/D layout: lane -> N = lane&15, VGPR v -> M = v + 8*(lane>>4)
#pragma unroll
    for (int v = 0; v < 8; ++v) {
        const int M = v + 8 * half;
        if (PASS == 0) {
            const int gi = gmap((tile * 16 + nloc) + (M << 8));
            lre[gi] = acc_re[v];
            lim[gi] = acc_im[v];
        } else if (PASS == 1) {
            const int gi = gmap(nloc + (M << 4) + (tile << 8));
            lre[gi] = acc_re[v];
            lim[gi] = acc_im[v];
        } else {
            const int q = M + (nloc << 4) + (tile << 8);
            const int f = q + (q >> 4) + (q >> 8);   // = M + 17*nloc + 273*tile
            v2f o; o.x = acc_re[v]; o.y = acc_im[v];
            lout[f] = o;
        }
    }
}

// ---- init kernel: digit-swapped twiddle table into d_ws -------------------
__global__ void twiddle_init_kernel(v2f* __restrict__ tw)
{
    const int s = blockIdx.x * blockDim.x + threadIdx.x;
    if (s < FFT_N) {
        const int e = ((s & 255) << 4) | (s >> 8);       // inverse of rho()
        float sn, cs;
        sincosf(-TWO_PI_F * (float)e * (1.0f / (float)FFT_N), &sn, &cs);
        v2f t; t.x = cs; t.y = sn;
        tw[s] = t;
    }
}

__global__ __launch_bounds__(256)
void fft4096_wmma_kernel(const float* __restrict__ x_re, const float* __restrict__ x_im,
                         float* __restrict__ y_re, float* __restrict__ y_im,
                         const v2f* __restrict__ tw_global, int use_ws)
{
    __shared__ float lre[4352];       // g-padded plane (max gmap(4095)=4350)
    __shared__ float lim[4352];
    __shared__ v2f   ltw[FFT_N];      // twiddles, rho() layout
    __shared__ v2f   lout[4368];      // pass-3 output, f() layout (max 4365)

    const int row  = blockIdx.x;
    const int tid  = threadIdx.x;
    const int lane = tid & 31;
    const int wave = tid >> 5;

    // ---- stage input row (g-padded via TDM pad feature) + twiddle table ----
    if (tid == 0) {
        const unsigned long long are =
            (unsigned long long)(uintptr_t)(x_re + (size_t)row * FFT_N);
        const unsigned long long aim =
            (unsigned long long)(uintptr_t)(x_im + (size_t)row * FFT_N);
        tdm_load_1d(are, (unsigned int)(uintptr_t)(&lre[0]), FFT_N, true);
        tdm_load_1d(aim, (unsigned int)(uintptr_t)(&lim[0]), FFT_N, true);
        if (use_ws) {
            tdm_load_1d((unsigned long long)(uintptr_t)tw_global,
                        (unsigned int)(uintptr_t)(&ltw[0]), 2 * FFT_N, false);
        }
        __builtin_amdgcn_s_wait_tensorcnt(0);
    }
    if (!use_ws) {   // fallback: build table in-kernel
        for (int s = tid; s < FFT_N; s += 256) {
            const int e = ((s & 255) << 4) | (s >> 8);
            float sn, cs;
            sincosf(-TWO_PI_F * (float)e * (1.0f / (float)FFT_N), &sn, &cs);
            v2f t; t.x = cs; t.y = sn;
            ltw[s] = t;
        }
    }
    __syncthreads();

    // ---- W16 A-fragments from the table: W[M][K] = w4096^(256*((M*K)&15)) ----
    // f32 16x4 A layout: lane&15 = M row; chunk cc, vgpr j, half h -> K = 4cc+2h+j
    const int half = lane >> 4;
    const int mrow = lane & 15;
    v2f wre[4], wim[4], wimn[4];
#pragma unroll
    for (int cc = 0; cc < 4; ++cc) {
#pragma unroll
        for (int j = 0; j < 2; ++j) {
            const int K = 4 * cc + 2 * half + j;
            const int m = (mrow * K) & 15;
            const v2f t = ltw[16 * m];            // rho(256*m) == 16*m
            wre[cc][j]  = t.x;
            wim[cc][j]  = t.y;
            wimn[cc][j] = -t.y;
        }
    }

    // ---- three radix-16 passes; 8 waves x 2 tiles cover 16 tiles ----
    fft_pass_tile<0>(lre, lim, ltw, lout, 2 * wave + 0, lane, wre, wim, wimn);
    fft_pass_tile<0>(lre, lim, ltw, lout, 2 * wave + 1, lane, wre, wim, wimn);
    __syncthreads();
    fft_pass_tile<1>(lre, lim, ltw, lout, 2 * wave + 0, lane, wre, wim, wimn);
    fft_pass_tile<1>(lre, lim, ltw, lout, 2 * wave + 1, lane, wre, wim, wimn);
    __syncthreads();
    fft_pass_tile<2>(lre, lim, ltw, lout, 2 * wave + 0, lane, wre, wim, wimn);
    fft_pass_tile<2>(lre, lim, ltw, lout, 2 * wave + 1, lane, wre, wim, wimn);
    __syncthreads();

    // ---- digit-reversed (base-16 digits 0<->2) coalesced store ----
    float* out_re = y_re + (size_t)row * FFT_N;
    float* out_im = y_im + (size_t)row * FFT_N;
    const int d1 = (tid >> 4) & 15;
    const int d2 = tid & 15;
#pragma unroll
    for (int i = 0; i < FFT_N / 256; ++i) {
        const int k = tid + 256 * i;              // consecutive lanes -> consecutive k
        const v2f v = lout[i + 17 * d1 + 273 * d2];
        out_re[k] = v.x;
        out_im[k] = v.y;
    }
}

extern "C" void kernel_launch(void* const* d_in, const int* in_sizes, int n_in,
                              void* d_out, int out_size, void* d_ws, size_t ws_size,
                              hipStream_t stream)
{
    const float* x_re = (const float*)d_in[0];
    const float* x_im = (const float*)d_in[1];
    const int rows = in_sizes[0] / FFT_N;         // 2048 for the reference shapes

    float* y_re = (float*)d_out;
    float* y_im = y_re + (size_t)rows * FFT_N;

    const int use_ws = (ws_size >= (size_t)FFT_N * sizeof(v2f)) ? 1 : 0;
    v2f* tw = (v2f*)d_ws;

    if (use_ws) {
        twiddle_init_kernel<<<FFT_N / 256, 256, 0, stream>>>(tw);
    }
    fft4096_wmma_kernel<<<rows, 256, 0, stream>>>(x_re, x_im, y_re, y_im, tw, use_ws);
}